// MyLoss_3891240370190
// MI455X (gfx1250) — compile-verified
//
#include <hip/hip_runtime.h>
#include <hip/hip_bf16.h>
#include <math.h>

typedef __attribute__((ext_vector_type(2))) float v2f;
typedef __attribute__((ext_vector_type(8))) float v8f;

#define NBOX 256
#define MPTS 512

__device__ __forceinline__ float sigmoidf_fast(float x) {
    return 1.0f / (1.0f + __expf(-x));
}

__global__ void zero_out_kernel(float* out) { out[0] = 0.0f; }

// ---------------------------------------------------------------------------
// Kernel 1: per (n,m) precompute guass_x, guass_y and interleaved
// (u,v) = (A1*grad_x, A1*grad_y)
// ---------------------------------------------------------------------------
__global__ __launch_bounds__(256)
void precompute_kernel(const float* __restrict__ pred,
                       const float* __restrict__ target,
                       const float* __restrict__ mask,
                       const float* __restrict__ grad_x,
                       const float* __restrict__ grad_y,
                       const float* __restrict__ offset,
                       float* __restrict__ gaussX,
                       float* __restrict__ gaussY,
                       float* __restrict__ uv) {
    int idx = blockIdx.x * blockDim.x + threadIdx.x;
    if (idx >= NBOX * MPTS) return;
    int n = idx >> 9;  // M = 512

    float ox = offset[2 * n + 0], oy = offset[2 * n + 1];
    float gxt = target[5 * n + 0] - ox;
    float gyt = target[5 * n + 1] - oy;
    float gw  = target[5 * n + 2];
    float gh  = target[5 * n + 3];
    float ga  = target[5 * n + 4];
    float px  = pred[5 * n + 0] - ox;
    float py  = pred[5 * n + 1] - oy;
    float pw  = pred[5 * n + 2];
    float ph  = pred[5 * n + 3];
    float pa  = pred[5 * n + 4];

    float mx = mask[2 * idx + 0];
    float my = mask[2 * idx + 1];

    float dx = mx - gxt, dy = my - gyt;
    float d  = sqrtf(dx * dx + dy * dy);
    float arg = (gxt - mx) / d;
    arg = fminf(1.0f, fmaxf(-1.0f, arg));
    float acv  = acosf(arg);
    float beta = (gyt >= my) ? (ga + acv) : (ga - acv);

    float sb, cb;
    __sincosf(beta, &sb, &cb);
    float d_w = fabsf(d * cb);
    float d_h = fabsf(d * sb);

    const float Kc = 15.0f;
    float a1 = (1.0f - sigmoidf_fast(Kc * (d_w - gw) / gw)) *
               (1.0f - sigmoidf_fast(Kc * (d_h - gh) / gh));

    float dwp = pw * d_w / gw;
    float dhp = ph * d_h / gh;

    float sa, ca;
    __sincosf(pa, &sa, &ca);
    float wx = dwp * ca, wy = dwp * sa;
    float hx = -dhp * sa, hy = dhp * ca;

    const float HPI = 1.57079632679489662f;
    const float PIF = 3.14159265358979323f;
    float gpx, gpy;
    if (beta > -HPI && beta <= 0.0f)        { gpx =  wx - hx; gpy =  wy - hy; }
    else if (beta > -PIF && beta <= -HPI)   { gpx =  wx + hx; gpy =  wy + hy; }
    else if (beta >  HPI && beta <=  PIF)   { gpx = -wx + hx; gpy = -wy + hy; }
    else                                    { gpx = -wx - hx; gpy = -wy - hy; }

    gaussX[idx] = px + gpx;
    gaussY[idx] = py + gpy;
    uv[2 * idx + 0] = a1 * grad_x[idx];
    uv[2 * idx + 1] = a1 * grad_y[idx];
}

// ---------------------------------------------------------------------------
// Kernel 2: per box n, Sigma_j A2[i,j]*u_j and Sigma_j A2[i,j]*v_j via
// V_WMMA_F32_16X16X4_F32; per-n j-data staged in LDS via async global->LDS.
// One wave = 16 i-rows; block = 8 waves; 4 blocks per box.
// ---------------------------------------------------------------------------
__global__ __launch_bounds__(256)
void gauss_wmma_kernel(const float* __restrict__ mask,
                       const float* __restrict__ grad_x,
                       const float* __restrict__ grad_y,
                       const float* __restrict__ gaussX,
                       const float* __restrict__ gaussY,
                       const float* __restrict__ uv,
                       float* __restrict__ out) {
    __shared__ __align__(16) float smxy[2 * MPTS];  // interleaved (mx,my), 4 KB
    __shared__ __align__(16) float suv[2 * MPTS];   // interleaved (u,v),   4 KB

    const int tid = threadIdx.x;
    const int n   = blockIdx.x >> 2;

    // ---- async stage of per-n data into LDS (ASYNCcnt path) ----
    {
        unsigned voff = (unsigned)tid * 16u;
        unsigned ldsA = (unsigned)(size_t)(&smxy[0]) + voff;
        unsigned ldsB = (unsigned)(size_t)(&suv[0]) + voff;
        unsigned long long gmask = (unsigned long long)(mask + (size_t)n * MPTS * 2);
        unsigned long long guv   = (unsigned long long)(uv   + (size_t)n * MPTS * 2);
        asm volatile("global_load_async_to_lds_b128 %0, %1, %2"
                     :: "v"(ldsA), "v"(voff), "s"(gmask) : "memory");
        asm volatile("global_load_async_to_lds_b128 %0, %1, %2"
                     :: "v"(ldsB), "v"(voff), "s"(guv) : "memory");
        asm volatile("s_wait_asynccnt 0" ::: "memory");
    }
    __syncthreads();

    const int lane  = tid & 31;
    const int wave  = tid >> 5;
    const int col   = lane & 15;            // N-column of WMMA this lane feeds
    const int khalf = (lane >> 4) << 1;     // lanes 0-15 -> K=0,1 ; 16-31 -> K=2,3
    const int i0    = ((blockIdx.x & 3) * 8 + wave) * 16;
    const int i     = i0 + col;

    const float gX = gaussX[(size_t)n * MPTS + i];
    const float gY = gaussY[(size_t)n * MPTS + i];

    // branchless B-column selectors (hoisted out of the loop)
    const float isU = (col == 0) ? 1.0f : 0.0f;
    const float isV = (col == 1) ? 1.0f : 0.0f;

    const float invC = 1.0f / (2.0f * 400.0f * 400.0f);  // 1/(2*THETA2^2)

    v8f c = {0.f, 0.f, 0.f, 0.f, 0.f, 0.f, 0.f, 0.f};

#pragma unroll 4
    for (int j0 = 0; j0 < MPTS; j0 += 4) {
        const int jb = j0 + khalf;  // this lane covers j = jb, jb+1
        // 16B-aligned b128 LDS loads: {mx_j, my_j, mx_j1, my_j1}, {u_j, v_j, u_j1, v_j1}
        float4 mq = *reinterpret_cast<const float4*>(&smxy[2 * jb]);
        float4 uq = *reinterpret_cast<const float4*>(&suv[2 * jb]);

        float dgx0 = gX - mq.x, dgy0 = gY - mq.y;
        float dgx1 = gX - mq.z, dgy1 = gY - mq.w;

        v2f a, b;
        a[0] = __expf((dgx0 * dgx0 + dgy0 * dgy0) * invC);
        a[1] = __expf((dgx1 * dgx1 + dgy1 * dgy1) * invC);
        b[0] = fmaf(isU, uq.x, isV * uq.y);
        b[1] = fmaf(isU, uq.z, isV * uq.w);

        c = __builtin_amdgcn_wmma_f32_16x16x4_f32(
                /*neg_a=*/false, a, /*neg_b=*/false, b,
                /*c_mod=*/(short)0, c, /*reuse_a=*/false, /*reuse_b=*/false);
    }

    // D column 0 = Sigma_j A2*u (per i), column 1 = Sigma_j A2*v.
    // C/D layout: VGPR r holds M=r (lanes 0-15) and M=r+8 (lanes 16-31).
    float acc = 0.0f;
    if (col < 2) {
        const float* g = (col == 0) ? grad_x : grad_y;
        int mbase = i0 + ((lane >> 4) << 3);
#pragma unroll
        for (int r = 0; r < 8; ++r)
            acc += c[r] * g[(size_t)n * MPTS + mbase + r];
    }
#pragma unroll
    for (int off = 16; off; off >>= 1)
        acc += __shfl_xor(acc, off, 32);

    if (lane == 0) {
        // fold in 1/(2*pi*THETA2), mean over (M,M), mean over N, LOSS_WEIGHT
        const float SCALE = (float)(1.0 / (2.0 * 3.14159265358979323846 * 400.0)
                                    / (512.0 * 512.0) / 256.0);
        atomicAdd(out, acc * SCALE);
    }
}

// ---------------------------------------------------------------------------
extern "C" void kernel_launch(void* const* d_in, const int* in_sizes, int n_in,
                              void* d_out, int out_size, void* d_ws, size_t ws_size,
                              hipStream_t stream) {
    const float* pred   = (const float*)d_in[0];
    const float* target = (const float*)d_in[1];
    const float* mask   = (const float*)d_in[2];
    const float* grad_x = (const float*)d_in[3];
    const float* grad_y = (const float*)d_in[4];
    const float* offset = (const float*)d_in[5];
    float* out = (float*)d_out;

    float* ws     = (float*)d_ws;
    float* gaussX = ws;                               // N*M floats
    float* gaussY = ws + (size_t)NBOX * MPTS;         // N*M floats
    float* uv     = ws + (size_t)2 * NBOX * MPTS;     // 2*N*M floats (interleaved)

    zero_out_kernel<<<1, 1, 0, stream>>>(out);

    precompute_kernel<<<(NBOX * MPTS) / 256, 256, 0, stream>>>(
        pred, target, mask, grad_x, grad_y, offset,
        gaussX, gaussY, uv);

    gauss_wmma_kernel<<<NBOX * 4, 256, 0, stream>>>(
        mask, grad_x, grad_y, gaussX, gaussY, uv, out);
}